// DySample_25563645346679
// MI455X (gfx1250) — compile-verified
//
#include <hip/hip_runtime.h>
#include <hip/hip_bf16.h>

typedef __attribute__((ext_vector_type(2))) float v2f;
typedef __attribute__((ext_vector_type(8))) float v8f;

#define IH 256
#define IW 256
#define IC 64
#define OH 512
#define OW 512

// One block handles a 16x16 pixel tile of one batch image.
// Phase A: 8 waves x 2 pixel-rows each -> WMMA f32 16x16x4, K=64 in 16 steps,
//          M-tiles {0..15}=x-offsets, {16..31}=y-offsets. Coords -> LDS.
// Phase B: 64 ch * 32 * 32 outputs, coalesced rows, bilinear gathers from L2.
__global__ __launch_bounds__(256) void dysample_fused(
    const float* __restrict__ x,      // (2, 64, 256, 256)
    const float* __restrict__ w_off,  // (32, 64)
    float* __restrict__ out)          // (2, 64, 512, 512)
{
    __shared__ float2 lds_c[4096];    // [gidx(4)][pix(256)][sub(4)] = 32 KB

    const int wg   = blockIdx.x;
    const int b    = wg >> 8;         // 512 blocks: 256 tiles per batch
    const int tile = wg & 255;
    const int ty   = tile >> 4;
    const int tx   = tile & 15;

    const int tid  = threadIdx.x;
    const int lane = tid & 31;
    const int wid  = tid >> 5;
    const int half = lane >> 4;       // 0: lanes 0-15, 1: lanes 16-31
    const int n    = lane & 15;

    // ---------------- Phase A: WMMA offset conv + coord build ----------------
    #pragma unroll
    for (int pgi = 0; pgi < 2; ++pgi) {
        const int pg  = wid * 2 + pgi;      // tile row index (pixel group)
        const int pix = pg * 16 + n;        // pixel id within tile
        const int gh  = ty * 16 + pg;       // global input row
        const int gw  = tx * 16 + n;        // global input col

        const float* xb = x + ((size_t)b * IC) * (IH * IW) + (size_t)gh * IW + gw;

        v8f accX = {};  // M-tile 0: w_off rows 0..15  (x offsets)
        v8f accY = {};  // M-tile 1: w_off rows 16..31 (y offsets)

        #pragma unroll
        for (int ks = 0; ks < 16; ++ks) {
            const int c0 = ks * 4 + half * 2;   // this lane's K pair
            v2f bf;
            bf.x = xb[(size_t)c0 * (IH * IW)];
            bf.y = xb[(size_t)(c0 + 1) * (IH * IW)];
            v2f aX = *(const v2f*)(w_off + n * IC + c0);
            v2f aY = *(const v2f*)(w_off + (16 + n) * IC + c0);
            accX = __builtin_amdgcn_wmma_f32_16x16x4_f32(
                       false, aX, false, bf, (short)0, accX, false, false);
            accY = __builtin_amdgcn_wmma_f32_16x16x4_f32(
                       false, aY, false, bf, (short)0, accY, false, false);
        }

        // D layout: VGPR j, lanes 0-15 -> M=j ; lanes 16-31 -> M=j+8.
        #pragma unroll
        for (int j = 0; j < 8; ++j) {
            const int m16 = j + half * 8;          // channel within axis (0..15)
            const int gi  = m16 >> 2;              // sampling group 0..3
            const int sub = m16 & 3;               // p*2+q sub-position
            float fx = (float)gw + 0.25f * accX[j] + ((sub & 1) ? 0.25f : -0.25f);
            float fy = (float)gh + 0.25f * accY[j] + ((sub & 2) ? 0.25f : -0.25f);
            fx = fminf(fmaxf(fx, 0.0f), (float)(IW - 1));
            fy = fminf(fmaxf(fy, 0.0f), (float)(IH - 1));
            lds_c[(gi * 256 + pix) * 4 + sub] = make_float2(fx, fy);
        }
    }

    __syncthreads();

    // ---------------- Phase B: bilinear sample + coalesced store -------------
    // 64 channels * 32 rows * 32 cols = 65536 outputs per block; 256/iter.
    #pragma unroll 4
    for (int it = 0; it < 256; ++it) {
        const int oidx = it * 256 + tid;
        const int x2l  = oidx & 31;
        const int y2l  = (oidx >> 5) & 31;
        const int Cc   = oidx >> 10;               // 0..63 (== gidx*16+cc)
        const int gi   = Cc >> 4;
        const int pix  = (y2l >> 1) * 16 + (x2l >> 1);
        const int sub  = (y2l & 1) * 2 + (x2l & 1);

        const float2 co = lds_c[(gi * 256 + pix) * 4 + sub];
        const float ix = co.x, iy = co.y;
        const float x0 = floorf(ix), y0 = floorf(iy);
        const float wx = ix - x0,   wy = iy - y0;
        const int x0i = (int)x0;
        const int y0i = (int)y0;
        const int x1i = min(x0i + 1, IW - 1);
        const int y1i = min(y0i + 1, IH - 1);

        const float* img = x + ((size_t)(b * IC + Cc)) * (IH * IW);
        const float v00 = img[y0i * IW + x0i];
        const float v01 = img[y0i * IW + x1i];
        const float v10 = img[y1i * IW + x0i];
        const float v11 = img[y1i * IW + x1i];

        const float top = v00 + (v01 - v00) * wx;
        const float bot = v10 + (v11 - v10) * wx;
        const float val = top + (bot - top) * wy;

        float* op = out + ((size_t)(b * IC + Cc)) * (OH * OW)
                        + (size_t)(ty * 32 + y2l) * OW + (tx * 32 + x2l);
        __builtin_nontemporal_store(val, op);   // write-once 134 MB stream
    }
}

extern "C" void kernel_launch(void* const* d_in, const int* in_sizes, int n_in,
                              void* d_out, int out_size, void* d_ws, size_t ws_size,
                              hipStream_t stream) {
    const float* x     = (const float*)d_in[0];   // (2,64,256,256)
    const float* w_off = (const float*)d_in[1];   // (32,64)
    float*       out   = (float*)d_out;           // (2,64,512,512)
    (void)in_sizes; (void)n_in; (void)out_size; (void)d_ws; (void)ws_size;

    dysample_fused<<<dim3(512), dim3(256), 0, stream>>>(x, w_off, out);
}